// RGCN_88845693485607
// MI455X (gfx1250) — compile-verified
//
#include <hip/hip_runtime.h>

typedef float v2f __attribute__((ext_vector_type(2)));
typedef float v8f __attribute__((ext_vector_type(8)));

#define ND 50000
#define EE 600000
#define DD 128
#define NB 4
#define NW 5   // 4 bases + self-loop weight

// ---------------------------------------------------------------------------
// Fused batched WMMA GEMM (fp32, V_WMMA_F32_16X16X4_F32).
// One kernel computes, for each 16-row node tile:
//   xb[:, b, :] = X @ V[b]   (b = 0..3)
//   hpre        = X @ Wl + bias
// Each wave owns one 16-col tile and 5 accumulators; the A fragment is loaded
// ONCE per k-step and reused by 5 WMMAs (5x less A traffic than grid.y=5).
// 8 waves/block cover all 128 columns. K=128 in steps of 4 -> 160 wmma/wave.
// Fragment layouts per CDNA5 ISA 7.12.2:
//   A (16x4 f32): lane<16: row=lane, k={k0,k0+1}; lane>=16: k={k0+2,k0+3}
//   B (4x16 f32): v0 = row k0+2*half, v1 = row k0+1+2*half, col = lane&15
//   C/D (16x16):  vgpr r -> (M = r + 8*half, N = col0 + (lane&15))
// ---------------------------------------------------------------------------
__global__ __launch_bounds__(256) void rgcn_gemm(
    const float* __restrict__ X,     // [N,128]
    const float* __restrict__ V,     // [4,128,128]
    const float* __restrict__ Wl,    // [128,128]
    const float* __restrict__ bias,  // [128]
    float* __restrict__ xb,          // [N,4,128]
    float* __restrict__ hpre)        // [N,128]
{
    const int wave = threadIdx.x >> 5;
    const int lane = threadIdx.x & 31;
    const int half = lane >> 4;      // 0 or 1
    const int m    = lane & 15;
    const int row0 = blockIdx.x * 16;
    const int col0 = wave * 16;

    const float* __restrict__ Wp[NW];
    Wp[0] = V;
    Wp[1] = V + 1 * DD * DD;
    Wp[2] = V + 2 * DD * DD;
    Wp[3] = V + 3 * DD * DD;
    Wp[4] = Wl;

    const float* __restrict__ Arow = X + (size_t)(row0 + m) * DD;

    v8f acc[NW];
    #pragma unroll
    for (int w = 0; w < NW; ++w) acc[w] = (v8f){};

    #pragma unroll 4
    for (int k0 = 0; k0 < DD; k0 += 4) {
        const int kk = k0 + 2 * half;
        const v2f a = *(const v2f*)(Arow + kk);        // 8B-aligned (kk even)
        #pragma unroll
        for (int w = 0; w < NW; ++w) {
            v2f b;
            b.x = Wp[w][(size_t)(kk    ) * DD + col0 + m];
            b.y = Wp[w][(size_t)(kk + 1) * DD + col0 + m];
            acc[w] = __builtin_amdgcn_wmma_f32_16x16x4_f32(
                        /*neg_a=*/false, a, /*neg_b=*/false, b,
                        /*c_mod=*/(short)0, acc[w],
                        /*reuse_a=*/false, /*reuse_b=*/false);
        }
    }

    // xb tiles (bases 0..3)
    #pragma unroll
    for (int w = 0; w < NB; ++w) {
        float* out = xb + (size_t)row0 * (NB * DD) + (size_t)w * DD + col0 + m;
        #pragma unroll
        for (int r = 0; r < 8; ++r) {
            const int rr = r + 8 * half;
            out[(size_t)rr * (NB * DD)] = acc[w][r];
        }
    }
    // self-loop tile + bias
    {
        const float bb = bias[col0 + m];
        float* out = hpre + (size_t)row0 * DD + col0 + m;
        #pragma unroll
        for (int r = 0; r < 8; ++r) {
            const int rr = r + 8 * half;
            out[(size_t)rr * DD] = acc[NB][r] + bb;
        }
    }
}

// ---------------------------------------------------------------------------
// Edge message + scatter-add. One wave per edge; lane owns 4 features.
// msg = sum_b comb[etype[e],b] * xb[src[e],b,:]; atomicAdd into hpre[dst[e],:]
// xb (102 MB) is L2-resident on MI455X (192 MB L2) -> gathers hit L2.
// ---------------------------------------------------------------------------
__global__ __launch_bounds__(256) void rgcn_edges(
    const float* __restrict__ xb,    // [N,4,128]
    const int*   __restrict__ src,
    const int*   __restrict__ dst,
    const int*   __restrict__ et,
    const float* __restrict__ comb,  // [R,4]
    float* __restrict__ hpre,        // [N,128]
    int nwaves_total)
{
    const int lane   = threadIdx.x & 31;
    const int wave_g = blockIdx.x * (blockDim.x >> 5) + (threadIdx.x >> 5);

    for (int e = wave_g; e < EE; e += nwaves_total) {
        const int s = src[e];
        const int d = dst[e];
        const int r = et[e];
        const float4 cf = *(const float4*)(comb + (size_t)r * 4);  // 16B aligned

        const float* xs = xb + (size_t)s * (NB * DD) + lane * 4;
        const float4 m0 = *(const float4*)(xs);
        const float4 m1 = *(const float4*)(xs + DD);
        const float4 m2 = *(const float4*)(xs + 2 * DD);
        const float4 m3 = *(const float4*)(xs + 3 * DD);

        float4 a;
        a.x = cf.x * m0.x + cf.y * m1.x + cf.z * m2.x + cf.w * m3.x;
        a.y = cf.x * m0.y + cf.y * m1.y + cf.z * m2.y + cf.w * m3.y;
        a.z = cf.x * m0.z + cf.y * m1.z + cf.z * m2.z + cf.w * m3.z;
        a.w = cf.x * m0.w + cf.y * m1.w + cf.z * m2.w + cf.w * m3.w;

        float* hp = hpre + (size_t)d * DD + lane * 4;
        unsafeAtomicAdd(hp + 0, a.x);    // -> global_atomic_add_f32 (no return)
        unsafeAtomicAdd(hp + 1, a.y);
        unsafeAtomicAdd(hp + 2, a.z);
        unsafeAtomicAdd(hp + 3, a.w);
    }
}

// ---------------------------------------------------------------------------
// BatchNorm: zero stats -> column sum/sumsq of relu(h) -> scale/shift -> apply
// ---------------------------------------------------------------------------
__global__ void rgcn_zero(float* __restrict__ stats)
{
    stats[threadIdx.x] = 0.0f;   // 256 threads: [0..127]=sum, [128..255]=sumsq
}

__global__ __launch_bounds__(256) void rgcn_stats(
    const float* __restrict__ hpre, float* __restrict__ stats)
{
    const int d  = threadIdx.x & 127;     // coalesced: consecutive tid -> consecutive col
    const int rh = threadIdx.x >> 7;      // 0/1: two row streams per block
    float s = 0.0f, ss = 0.0f;
    for (int n = blockIdx.x * 2 + rh; n < ND; n += gridDim.x * 2) {
        float v = hpre[(size_t)n * DD + d];
        v = fmaxf(v, 0.0f);               // ReLU precedes BN in the reference
        s += v;
        ss += v * v;
    }
    __shared__ float sh[512];
    sh[threadIdx.x]       = s;
    sh[256 + threadIdx.x] = ss;
    __syncthreads();
    if (threadIdx.x < 128) {
        s  = sh[threadIdx.x]       + sh[threadIdx.x + 128];
        ss = sh[256 + threadIdx.x] + sh[384 + threadIdx.x];
        unsafeAtomicAdd(&stats[d],       s);
        unsafeAtomicAdd(&stats[128 + d], ss);
    }
}

__global__ void rgcn_finalize(
    const float* __restrict__ stats,
    const float* __restrict__ gamma,
    const float* __restrict__ beta,
    float* __restrict__ sc)              // [0..127]=scale, [128..255]=shift
{
    const int d = threadIdx.x;           // 128 threads
    const float inv_n = 1.0f / (float)ND;
    const float mu  = stats[d] * inv_n;
    const float var = stats[128 + d] * inv_n - mu * mu;   // biased var
    const float rs  = rsqrtf(var + 1e-5f);
    const float scale = gamma[d] * rs;
    sc[d]       = scale;
    sc[128 + d] = beta[d] - mu * scale;
}

__global__ __launch_bounds__(256) void rgcn_apply(
    const float* __restrict__ hpre,
    const float* __restrict__ sc,
    float* __restrict__ out)
{
    const int i = blockIdx.x * 256 + threadIdx.x;   // grid sized exactly N*D/256
    const int d = i & 127;
    const float v = fmaxf(hpre[i], 0.0f);
    out[i] = v * sc[d] + sc[128 + d];
}

// ---------------------------------------------------------------------------
// Driver. Input order: feat, src, dst, etype, {V,comb,loop,bias,gamma,beta}x2
// ws layout: xb [N*4*128] | hpre [N*128] | stats [256] | sc [256]
// Layer-0 output lives in d_out; layer 1 reads it, then overwrites it.
// ---------------------------------------------------------------------------
extern "C" void kernel_launch(void* const* d_in, const int* in_sizes, int n_in,
                              void* d_out, int out_size, void* d_ws, size_t ws_size,
                              hipStream_t stream)
{
    (void)in_sizes; (void)n_in; (void)out_size; (void)ws_size;

    const float* feat = (const float*)d_in[0];
    const int*   src  = (const int*)  d_in[1];
    const int*   dst  = (const int*)  d_in[2];
    const int*   et   = (const int*)  d_in[3];

    float* out   = (float*)d_out;
    float* xb    = (float*)d_ws;                       // 25,600,000 f
    float* hpre  = xb   + (size_t)ND * NB * DD;        //  6,400,000 f
    float* stats = hpre + (size_t)ND * DD;             //        256 f
    float* sc    = stats + 256;                        //        256 f

    const int gemm_blocks  = ND / 16;                  // 3125
    const int edge_blocks  = 2048;
    const int edge_waves   = edge_blocks * 8;
    const int stat_blocks  = 256;
    const int apply_blocks = (ND * DD) / 256;          // exact: 25000

    for (int l = 0; l < 2; ++l) {
        const float* X     = (l == 0) ? feat : out;
        const float* V     = (const float*)d_in[4 + 6 * l];
        const float* comb  = (const float*)d_in[5 + 6 * l];
        const float* Wl    = (const float*)d_in[6 + 6 * l];
        const float* bias  = (const float*)d_in[7 + 6 * l];
        const float* gamma = (const float*)d_in[8 + 6 * l];
        const float* beta  = (const float*)d_in[9 + 6 * l];

        rgcn_gemm    <<<gemm_blocks, 256, 0, stream>>>(X, V, Wl, bias, xb, hpre);
        rgcn_edges   <<<edge_blocks, 256, 0, stream>>>(xb, src, dst, et, comb, hpre, edge_waves);
        rgcn_zero    <<<1, 256, 0, stream>>>(stats);
        rgcn_stats   <<<stat_blocks, 256, 0, stream>>>(hpre, stats);
        rgcn_finalize<<<1, 128, 0, stream>>>(stats, gamma, beta, sc);
        rgcn_apply   <<<apply_blocks, 256, 0, stream>>>(hpre, sc, out);
    }
}